// LSTM_54932631716304
// MI455X (gfx1250) — compile-verified
//
#include <hip/hip_runtime.h>
#include <math.h>

#define B_  128
#define S_  512
#define I_  256
#define H_  512
#define G4H (4 * H_)

typedef __attribute__((ext_vector_type(2))) float v2f;
typedef __attribute__((ext_vector_type(8))) float v8f;

__device__ __forceinline__ v2f ld2(const float* p) { return *(const v2f*)p; }
__device__ __forceinline__ float sigm(float x) { return 1.0f / (1.0f + __expf(-x)); }

#define WMMA_F32(ACC, A, Bf) \
    (ACC) = __builtin_amdgcn_wmma_f32_16x16x4_f32(false, (A), false, (Bf), (short)0, (ACC), false, false)

// ---------------------------------------------------------------------------
// Full-chip input projection: xg[row, g] = x_row . W_ih[g,:] + bias[g]
// rows = t*B + b  (S*B rows), cols = 4H. Each wave: 16 rows x (4 gates x 16).
// grid = (H/16, S*B/(16*8)), block = 256 (8 waves).
// ---------------------------------------------------------------------------
__global__ __launch_bounds__(256) void xg_proj_kernel(
    const float* __restrict__ x,      // (B, S, I)
    const float* __restrict__ W_ih,   // (4H, I)
    const float* __restrict__ b_ih,   // (4H)
    const float* __restrict__ b_hh,   // (4H)
    float* __restrict__ xg)           // (S*B, 4H)
{
    const int lane = threadIdx.x & 31;
    const int w    = threadIdx.x >> 5;
    const int lo   = lane & 15;
    const int hi   = lane >> 4;
    const int m0   = (blockIdx.y * 8 + w) * 16;   // row-tile base (row = t*B+b)
    const int n0   = blockIdx.x * 16;
    const int n    = n0 + lo;

    const int row = m0 + lo;
    const int tt  = row / B_;
    const int bb  = row % B_;
    const float* arow = x + (size_t)bb * ((size_t)S_ * I_) + (size_t)tt * I_;
    const float* w0 = W_ih + (size_t)(0 * H_ + n) * I_;
    const float* w1 = W_ih + (size_t)(1 * H_ + n) * I_;
    const float* w2 = W_ih + (size_t)(2 * H_ + n) * I_;
    const float* w3 = W_ih + (size_t)(3 * H_ + n) * I_;

    v8f acc0 = {}, acc1 = {}, acc2 = {}, acc3 = {};
    for (int k0 = 0; k0 < I_; k0 += 4) {
        const int kk = k0 + 2 * hi;
        v2f a  = ld2(arow + kk);
        v2f f0 = ld2(w0 + kk);
        v2f f1 = ld2(w1 + kk);
        v2f f2 = ld2(w2 + kk);
        v2f f3 = ld2(w3 + kk);
        WMMA_F32(acc0, a, f0);
        WMMA_F32(acc1, a, f1);
        WMMA_F32(acc2, a, f2);
        WMMA_F32(acc3, a, f3);
    }

    const float bias0 = b_ih[0 * H_ + n] + b_hh[0 * H_ + n];
    const float bias1 = b_ih[1 * H_ + n] + b_hh[1 * H_ + n];
    const float bias2 = b_ih[2 * H_ + n] + b_hh[2 * H_ + n];
    const float bias3 = b_ih[3 * H_ + n] + b_hh[3 * H_ + n];

#pragma unroll
    for (int r = 0; r < 8; ++r) {
        const size_t rowr = (size_t)(m0 + r + 8 * hi) * G4H;
        xg[rowr + 0 * H_ + n] = acc0[r] + bias0;
        xg[rowr + 1 * H_ + n] = acc1[r] + bias1;
        xg[rowr + 2 * H_ + n] = acc2[r] + bias2;
        xg[rowr + 3 * H_ + n] = acc3[r] + bias3;
    }
}

// ---------------------------------------------------------------------------
// Recurrent step with precomputed xg: gates = xg[t] + h @ W_hh^T  (K = H).
// One wave per workgroup to spread the small GEMM over many WGPs.
// grid = (H/16, B/16), block = 32.
// ---------------------------------------------------------------------------
__global__ __launch_bounds__(32) void lstm_step_pre_kernel(
    const float* __restrict__ xg,     // (S*B, 4H)
    const float* __restrict__ W_hh,   // (4H, H)
    const float* __restrict__ h_in,   // (B, H)
    float* __restrict__ h_out,        // (B, H)
    float* __restrict__ c_state,      // (B, H)
    int t)
{
    const int lane = threadIdx.x & 31;
    const int lo   = lane & 15;
    const int hi   = lane >> 4;
    const int m0   = blockIdx.y * 16;
    const int n0   = blockIdx.x * 16;
    const int n    = n0 + lo;

    v8f acc0 = {}, acc1 = {}, acc2 = {}, acc3 = {};
    {
        const float* arow = h_in + (size_t)(m0 + lo) * H_;
        const float* w0 = W_hh + (size_t)(0 * H_ + n) * H_;
        const float* w1 = W_hh + (size_t)(1 * H_ + n) * H_;
        const float* w2 = W_hh + (size_t)(2 * H_ + n) * H_;
        const float* w3 = W_hh + (size_t)(3 * H_ + n) * H_;
        for (int k0 = 0; k0 < H_; k0 += 4) {
            const int kk = k0 + 2 * hi;
            v2f a  = ld2(arow + kk);
            v2f f0 = ld2(w0 + kk);
            v2f f1 = ld2(w1 + kk);
            v2f f2 = ld2(w2 + kk);
            v2f f3 = ld2(w3 + kk);
            WMMA_F32(acc0, a, f0);
            WMMA_F32(acc1, a, f1);
            WMMA_F32(acc2, a, f2);
            WMMA_F32(acc3, a, f3);
        }
    }

#pragma unroll
    for (int r = 0; r < 8; ++r) {
        const int m = m0 + r + 8 * hi;
        const size_t idx  = (size_t)m * H_ + n;
        const size_t rowr = (size_t)(t * B_ + m) * G4H;
        const float ig = sigm(acc0[r] + xg[rowr + 0 * H_ + n]);
        const float fg = sigm(acc1[r] + xg[rowr + 1 * H_ + n]);
        const float gg = tanhf(acc2[r] + xg[rowr + 2 * H_ + n]);
        const float og = sigm(acc3[r] + xg[rowr + 3 * H_ + n]);
        const float c_old = c_state[idx];
        const float c_new = fg * c_old + ig * gg;
        c_state[idx] = c_new;
        h_out[idx]   = og * tanhf(c_new);
    }
}

// ---------------------------------------------------------------------------
// Fallback fused step (no xg workspace): gates = x_t@W_ih^T + h@W_hh^T + bias.
// grid = (H/16, B/16), block = 32.
// ---------------------------------------------------------------------------
__global__ __launch_bounds__(32) void lstm_step_fused_kernel(
    const float* __restrict__ x, const float* __restrict__ W_ih,
    const float* __restrict__ W_hh, const float* __restrict__ b_ih,
    const float* __restrict__ b_hh, const float* __restrict__ h_in,
    float* __restrict__ h_out, float* __restrict__ c_state, int t)
{
    const int lane = threadIdx.x & 31;
    const int lo   = lane & 15;
    const int hi   = lane >> 4;
    const int m0   = blockIdx.y * 16;
    const int n0   = blockIdx.x * 16;
    const int n    = n0 + lo;

    v8f acc0 = {}, acc1 = {}, acc2 = {}, acc3 = {};
    {
        const float* arow = x + (size_t)(m0 + lo) * ((size_t)S_ * I_) + (size_t)t * I_;
        const float* w0 = W_ih + (size_t)(0 * H_ + n) * I_;
        const float* w1 = W_ih + (size_t)(1 * H_ + n) * I_;
        const float* w2 = W_ih + (size_t)(2 * H_ + n) * I_;
        const float* w3 = W_ih + (size_t)(3 * H_ + n) * I_;
        for (int k0 = 0; k0 < I_; k0 += 4) {
            const int kk = k0 + 2 * hi;
            v2f a  = ld2(arow + kk);
            v2f f0 = ld2(w0 + kk);
            v2f f1 = ld2(w1 + kk);
            v2f f2 = ld2(w2 + kk);
            v2f f3 = ld2(w3 + kk);
            WMMA_F32(acc0, a, f0);
            WMMA_F32(acc1, a, f1);
            WMMA_F32(acc2, a, f2);
            WMMA_F32(acc3, a, f3);
        }
    }
    {
        const float* arow = h_in + (size_t)(m0 + lo) * H_;
        const float* w0 = W_hh + (size_t)(0 * H_ + n) * H_;
        const float* w1 = W_hh + (size_t)(1 * H_ + n) * H_;
        const float* w2 = W_hh + (size_t)(2 * H_ + n) * H_;
        const float* w3 = W_hh + (size_t)(3 * H_ + n) * H_;
        for (int k0 = 0; k0 < H_; k0 += 4) {
            const int kk = k0 + 2 * hi;
            v2f a  = ld2(arow + kk);
            v2f f0 = ld2(w0 + kk);
            v2f f1 = ld2(w1 + kk);
            v2f f2 = ld2(w2 + kk);
            v2f f3 = ld2(w3 + kk);
            WMMA_F32(acc0, a, f0);
            WMMA_F32(acc1, a, f1);
            WMMA_F32(acc2, a, f2);
            WMMA_F32(acc3, a, f3);
        }
    }

    const float bias0 = b_ih[0 * H_ + n] + b_hh[0 * H_ + n];
    const float bias1 = b_ih[1 * H_ + n] + b_hh[1 * H_ + n];
    const float bias2 = b_ih[2 * H_ + n] + b_hh[2 * H_ + n];
    const float bias3 = b_ih[3 * H_ + n] + b_hh[3 * H_ + n];

#pragma unroll
    for (int r = 0; r < 8; ++r) {
        const int m = m0 + r + 8 * hi;
        const size_t idx = (size_t)m * H_ + n;
        const float ig = sigm(acc0[r] + bias0);
        const float fg = sigm(acc1[r] + bias1);
        const float gg = tanhf(acc2[r] + bias2);
        const float og = sigm(acc3[r] + bias3);
        const float c_old = c_state[idx];
        const float c_new = fg * c_old + ig * gg;
        c_state[idx] = c_new;
        h_out[idx]   = og * tanhf(c_new);
    }
}

// ---------------------------------------------------------------------------
// attn_W transpose (so attention B-fragments are contiguous b64 loads).
// grid = (H/16, H/16), block = 256 (16x16 tile).
// ---------------------------------------------------------------------------
__global__ __launch_bounds__(256) void transpose_kernel(
    const float* __restrict__ in, float* __restrict__ out)
{
    __shared__ float tile[16][17];
    const int tx = threadIdx.x & 15;
    const int ty = threadIdx.x >> 4;
    const int bx = blockIdx.x * 16;
    const int by = blockIdx.y * 16;
    tile[ty][tx] = in[(size_t)(by + ty) * H_ + bx + tx];
    __syncthreads();
    out[(size_t)(bx + ty) * H_ + by + tx] = tile[tx][ty];
}

// ---------------------------------------------------------------------------
// e[sb] = attn_v . tanh(hs[sb,:] @ attn_W + attn_b), using transposed attn_W.
// grid = S*B/(16*8), block = 256 (8 waves).
// ---------------------------------------------------------------------------
__global__ __launch_bounds__(256) void attn_e_kernel(
    const float* __restrict__ hs,       // (S*B, H)
    const float* __restrict__ attn_WT,  // (H, H) transposed: WT[j,h] = W[h,j]
    const float* __restrict__ attn_b,   // (H)
    const float* __restrict__ attn_v,   // (H)
    float* __restrict__ e)              // (S*B)
{
    const int lane = threadIdx.x & 31;
    const int w    = threadIdx.x >> 5;
    const int lo   = lane & 15;
    const int hi   = lane >> 4;
    const int m0   = (blockIdx.x * 8 + w) * 16;

    const float* arow = hs + (size_t)(m0 + lo) * H_;

    float ep[8];
#pragma unroll
    for (int r = 0; r < 8; ++r) ep[r] = 0.0f;

    for (int jt = 0; jt < H_; jt += 16) {
        const int j = jt + lo;
        const float* brow = attn_WT + (size_t)j * H_;
        v8f acc = {};
        for (int k0 = 0; k0 < H_; k0 += 4) {
            const int kk = k0 + 2 * hi;
            v2f a  = ld2(arow + kk);
            v2f bf = ld2(brow + kk);
            WMMA_F32(acc, a, bf);
        }
        const float bj = attn_b[j];
        const float vj = attn_v[j];
#pragma unroll
        for (int r = 0; r < 8; ++r)
            ep[r] += tanhf(acc[r] + bj) * vj;
    }

#pragma unroll
    for (int r = 0; r < 8; ++r) {
        float v = ep[r];
        v += __shfl_xor(v, 1, 32);
        v += __shfl_xor(v, 2, 32);
        v += __shfl_xor(v, 4, 32);
        v += __shfl_xor(v, 8, 32);
        ep[r] = v;
    }
    if (lo == 0) {
#pragma unroll
        for (int r = 0; r < 8; ++r)
            e[m0 + r + 8 * hi] = ep[r];
    }
}

// ---------------------------------------------------------------------------
__global__ __launch_bounds__(256) void softmax_kernel(float* __restrict__ e)
{
    __shared__ float red[256];
    const int b   = blockIdx.x;
    const int tid = threadIdx.x;

    const float v0 = e[(size_t)tid * B_ + b];
    const float v1 = e[(size_t)(tid + 256) * B_ + b];

    red[tid] = fmaxf(v0, v1);
    __syncthreads();
    for (int s2 = 128; s2 > 0; s2 >>= 1) {
        if (tid < s2) red[tid] = fmaxf(red[tid], red[tid + s2]);
        __syncthreads();
    }
    const float mx = red[0];
    __syncthreads();

    const float e0 = __expf(v0 - mx);
    const float e1 = __expf(v1 - mx);
    red[tid] = e0 + e1;
    __syncthreads();
    for (int s2 = 128; s2 > 0; s2 >>= 1) {
        if (tid < s2) red[tid] += red[tid + s2];
        __syncthreads();
    }
    const float inv = 1.0f / red[0];

    e[(size_t)tid * B_ + b]         = e0 * inv;
    e[(size_t)(tid + 256) * B_ + b] = e1 * inv;
}

// ---------------------------------------------------------------------------
__global__ __launch_bounds__(256) void context_kernel(
    const float* __restrict__ hs, const float* __restrict__ alpha,
    float* __restrict__ out)
{
    const int idx = blockIdx.x * 256 + threadIdx.x;   // b*H + h
    const int b = idx / H_;
    const int h = idx % H_;
    float acc = 0.0f;
    for (int s = 0; s < S_; ++s)
        acc += alpha[(size_t)s * B_ + b] * hs[((size_t)s * B_ + b) * H_ + h];
    out[idx] = acc;
}

__global__ void zero_kernel(float* p, int n)
{
    const int i = blockIdx.x * 256 + threadIdx.x;
    if (i < n) p[i] = 0.0f;
}

// ---------------------------------------------------------------------------
extern "C" void kernel_launch(void* const* d_in, const int* in_sizes, int n_in,
                              void* d_out, int out_size, void* d_ws, size_t ws_size,
                              hipStream_t stream)
{
    (void)in_sizes; (void)n_in; (void)out_size;

    const float* x      = (const float*)d_in[0];
    const float* W_ih   = (const float*)d_in[1];
    const float* W_hh   = (const float*)d_in[2];
    const float* b_ih   = (const float*)d_in[3];
    const float* b_hh   = (const float*)d_in[4];
    const float* attn_W = (const float*)d_in[5];
    const float* attn_b = (const float*)d_in[6];
    const float* attn_v = (const float*)d_in[7];
    float* out = (float*)d_out;

    // workspace layout (floats)
    const size_t hs_sz = (size_t)S_ * B_ * H_;       // 33.5M
    const size_t xg_sz = (size_t)S_ * B_ * G4H;      // 134M
    float* hs = (float*)d_ws;
    float* c  = hs + hs_sz;                          // B*H
    float* h0 = c  + (size_t)B_ * H_;                // B*H
    float* e  = h0 + (size_t)B_ * H_;                // S*B
    float* wt = e  + (size_t)S_ * B_;                // H*H
    float* xg = wt + (size_t)H_ * H_;                // S*B*4H (optional)

    const size_t base_floats = hs_sz + 2 * (size_t)B_ * H_ + (size_t)S_ * B_ + (size_t)H_ * H_;
    const bool use_xg = ws_size >= (base_floats + xg_sz) * sizeof(float);

    // zero c and h0 (contiguous)
    zero_kernel<<<(2 * B_ * H_ + 255) / 256, 256, 0, stream>>>(c, 2 * B_ * H_);
    // transpose attn_W once
    transpose_kernel<<<dim3(H_ / 16, H_ / 16), 256, 0, stream>>>(attn_W, wt);

    if (use_xg) {
        // full-chip input projection off the sequential critical path
        xg_proj_kernel<<<dim3(H_ / 16, (S_ * B_) / (16 * 8)), 256, 0, stream>>>(
            x, W_ih, b_ih, b_hh, xg);
        for (int t = 0; t < S_; ++t) {
            const float* h_in = (t == 0) ? h0 : hs + (size_t)(t - 1) * B_ * H_;
            float* h_out = hs + (size_t)t * B_ * H_;
            lstm_step_pre_kernel<<<dim3(H_ / 16, B_ / 16), 32, 0, stream>>>(
                xg, W_hh, h_in, h_out, c, t);
        }
    } else {
        for (int t = 0; t < S_; ++t) {
            const float* h_in = (t == 0) ? h0 : hs + (size_t)(t - 1) * B_ * H_;
            float* h_out = hs + (size_t)t * B_ * H_;
            lstm_step_fused_kernel<<<dim3(H_ / 16, B_ / 16), 32, 0, stream>>>(
                x, W_ih, W_hh, b_ih, b_hh, h_in, h_out, c, t);
        }
    }

    // attention
    attn_e_kernel<<<(S_ * B_) / (16 * 8), 256, 0, stream>>>(hs, wt, attn_b, attn_v, e);
    softmax_kernel<<<B_, 256, 0, stream>>>(e);
    context_kernel<<<(B_ * H_) / 256, 256, 0, stream>>>(hs, e, out);
}